// TensorSquare_53747220742303
// MI455X (gfx1250) — compile-verified
//
#include <hip/hip_runtime.h>

typedef __attribute__((ext_vector_type(2))) float v2f;
typedef __attribute__((ext_vector_type(8))) float v8f;

#define IN_DIM    480
#define OUT_DIM   115440
#define NVALU     39664
#define CG_FLOATS 966

// ======================= CG (real Wigner-3j) builder ========================
__device__ double dfact_(int n){ double r=1.0; for(int i=2;i<=n;++i) r*=(double)i; return r; }

__device__ double su2cg_(int j1,int m1,int j2,int m2,int j3,int m3){
  if (m1+m2 != m3) return 0.0;
  double pref = sqrt((double)(2*j3+1)
      * dfact_(j3+j1-j2)*dfact_(j3-j1+j2)*dfact_(j1+j2-j3)/dfact_(j1+j2+j3+1)
      * dfact_(j3+m3)*dfact_(j3-m3)*dfact_(j1-m1)*dfact_(j1+m1)*dfact_(j2-m2)*dfact_(j2+m2));
  int vmin = 0; if (j2-j3-m1>vmin) vmin=j2-j3-m1; if (j1-j3+m2>vmin) vmin=j1-j3+m2;
  int vmax = j1+j2-j3; if (j1-m1<vmax) vmax=j1-m1; if (j2+m2<vmax) vmax=j2+m2;
  double s=0.0;
  for(int v=vmin; v<=vmax; ++v){
    double t = 1.0/(dfact_(v)*dfact_(j1+j2-j3-v)*dfact_(j1-m1-v)*dfact_(j2+m2-v)
                    *dfact_(j3-j2+m1+v)*dfact_(j3-j1-m2+v));
    s += (v&1)? -t : t;
  }
  return pref*s;
}

__device__ void qmat_(int l, double qr[9][9], double qi[9][9]){
  for(int i=0;i<9;++i) for(int j=0;j<9;++j){ qr[i][j]=0.0; qi[i][j]=0.0; }
  double is2 = sqrt(0.5);
  for(int m=-l;m<0;++m){ qr[l+m][l-m]=is2; qi[l+m][l+m]=-is2; }
  qr[l][l]=1.0;
  for(int m=1;m<=l;++m){ double s=(m&1)?-1.0:1.0; qr[l+m][l+m]=s*is2; qi[l+m][l-m]=s*is2; }
  double fr,fi;                               // (-i)^l
  switch(l&3){ case 0: fr=1; fi=0; break; case 1: fr=0; fi=-1; break;
               case 2: fr=-1; fi=0; break; default: fr=0; fi=1; }
  int d=2*l+1;
  for(int i=0;i<d;++i)for(int j=0;j<d;++j){
    double r = qr[i][j]*fr - qi[i][j]*fi;
    double m = qr[i][j]*fi + qi[i][j]*fr;
    qr[i][j]=r; qi[i][j]=m;
  }
}

// stored[j][l][m] = (wigner3j(l1,l2,l3)/||.||_F) * sqrt(2*l3+1)
__global__ void init_cg_kernel(float* __restrict__ ws){
  const int L1[14]={0,0,0,1,1,1,1,1,1,2,2,2,2,2};
  const int L2[14]={0,1,2,1,1,1,2,2,2,2,2,2,2,2};
  const int L3[14]={0,1,2,0,1,2,1,2,3,0,1,2,3,4};
  const int OFF[14]={0,1,10,35,44,71,116,161,236,341,366,441,566,741};
  int t = threadIdx.x;
  if (t >= 14) return;
  int l1=L1[t], l2=L2[t], l3=L3[t], off=OFF[t];
  int d1=2*l1+1, d2=2*l2+1, d3=2*l3+1;
  double q1r[9][9],q1i[9][9],q2r[9][9],q2i[9][9],q3r[9][9],q3i[9][9];
  qmat_(l1,q1r,q1i); qmat_(l2,q2r,q2i); qmat_(l3,q3r,q3i);
  double C[5][5][9];
  for(int i=0;i<5;++i)for(int k=0;k<5;++k)for(int n=0;n<9;++n) C[i][k][n]=0.0;
  for(int i=0;i<d1;++i)for(int k=0;k<d2;++k){
    int m1=i-l1, m2=k-l2, m3=m1+m2;
    if (m3>=-l3 && m3<=l3) C[i][k][l3+m3]=su2cg_(l1,m1,l2,m2,l3,m3);
  }
  double W[5][5][9]; double nrm=0.0;
  for(int j=0;j<d1;++j)for(int l=0;l<d2;++l)for(int m=0;m<d3;++m){
    double acc=0.0;
    for(int i=0;i<d1;++i)for(int k=0;k<d2;++k)for(int n=0;n<d3;++n){
      double c=C[i][k][n]; if(c==0.0) continue;
      double t1r=q1r[i][j]*q2r[k][l]-q1i[i][j]*q2i[k][l];
      double t1i=q1r[i][j]*q2i[k][l]+q1i[i][j]*q2r[k][l];
      acc += c*(t1r*q3r[n][m] + t1i*q3i[n][m]);   // Re(q1*q2*conj(q3))*C
    }
    W[j][l][m]=acc; nrm+=acc*acc;
  }
  double sc = sqrt((double)d3/nrm);
  for(int j=0;j<d1;++j)for(int l=0;l<d2;++l)for(int m=0;m<d3;++m)
    ws[off + (j*d2+l)*d3 + m] = (float)(W[j][l][m]*sc);
}

// =========================== WMMA kernel ====================================
// Chained rank-1 update: D += a_col(16) x b_row(16) via V_WMMA_F32_16X16X4_F32.
// Only K=0 is used; K=0 of A and B fragments is VGPR0 / lanes 0-15 (ISA 7.12.2).
__device__ __forceinline__ v8f wmma_rank1(float aval, float bval, v8f c){
  v2f a; a.x = aval; a.y = 0.0f;
  v2f b; b.x = bval; b.y = 0.0f;
  return __builtin_amdgcn_wmma_f32_16x16x4_f32(false, a, false, b, (short)0, c, false, false);
}

// Store one 16x16 f32 C/D tile; N is a compile-time row stride so the backend
// folds r*N into the global_store immediate offset. Non-temporal: output is
// streamed once and exceeds L2.
template<int N>
__device__ __forceinline__ void store_tile(float* __restrict__ ob, int outoff,
                                           int mt, int n, bool lo16, v8f d){
  float* p = ob + outoff + (size_t)(mt*16 + (lo16?0:8))*N + n;
  #pragma unroll
  for(int r=0;r<8;++r) __builtin_nontemporal_store(d[r], p + r*N);
}

template<int N, int TOFF, int OUTOFF>
__device__ __forceinline__ void gemm_tile(const float* __restrict__ lx,
                                          const float* __restrict__ lT,
                                          float* __restrict__ ob,
                                          int mt, int nt, int lane15, bool lo16){
  int n  = nt*16 + lane15;
  int uu = mt*16 + lane15;
  v8f d = {0.f,0.f,0.f,0.f,0.f,0.f,0.f,0.f};
  #pragma unroll
  for(int i=0;i<3;++i)
    d = wmma_rank1(lo16 ? lx[128+uu*3+i] : 0.f,
                   lo16 ? lT[TOFF+i*N+n] : 0.f, d);
  store_tile<N>(ob, OUTOFF, mt, n, lo16, d);
}

// Covers: p2 (0e x 1o outer, 128x192 @ out 10864), p3 (0e x 2e outer, 128x160 @ 49120),
//         p9/p10/p11 (1o x 2e uvuv GEMMs, 64 x {96,160,224} @ {35440,82640,92880}).
__global__ __launch_bounds__(256) void tp_wmma_kernel(const float* __restrict__ x,
    const float* __restrict__ cgtab, float* __restrict__ out){
  __shared__ float lx[IN_DIM];
  __shared__ float lcg[CG_FLOATS];
  __shared__ float lT[1440];          // T for lo=1(3x96) | lo=2(3x160) | lo=3(3x224)
  const int b = blockIdx.x;
  for (int i=threadIdx.x;i<IN_DIM;i+=256)    lx[i]=x[(size_t)b*IN_DIM+i];
  for (int i=threadIdx.x;i<CG_FLOATS;i+=256) lcg[i]=cgtab[i];
  __syncthreads();
  // phase 1: T[i][v*dk+k] = sum_j cg(1,2,lo)[i][j][k] * x2[v][j]
  for (int tt=threadIdx.x; tt<1440; tt+=256){
    int Toff,N,dk,cgo;
    if (tt<288){Toff=0;N=96;dk=3;cgo=116;}
    else if (tt<768){Toff=288;N=160;dk=5;cgo=161;}
    else {Toff=768;N=224;dk=7;cgo=236;}
    int local=tt-Toff; int i=local/N; int n=local-i*N;
    int v=n/dk; int k=n-v*dk;
    float acc=0.f;
    #pragma unroll
    for(int j=0;j<5;++j) acc += lcg[cgo+(i*5+j)*dk+k]*lx[320+v*5+j];
    lT[tt]=acc;
  }
  __syncthreads();
  // Scalarize the wave id so tile decode is SALU-only and EXEC stays all-ones
  // at every WMMA (hardware requirement, ISA 7.12).
  const int wave   = __builtin_amdgcn_readfirstlane((int)(threadIdx.x >> 5));
  const int lane   = threadIdx.x & 31;
  const int lane15 = lane & 15;
  const bool lo16  = lane < 16;
  float* ob = out + (size_t)b*OUT_DIM;
  // 296 16x16 tiles: 96 (p2) + 80 (p3) + 24+40+56 (p9..p11); wave-uniform loop.
  for (int t=wave; t<296; t+=8){
    if (t < 96){                                   // p2: x0 (x) x1flat, N=192
      int mt=t/12, nt=t-mt*12;
      int n = nt*16+lane15;
      v8f d = {0.f,0.f,0.f,0.f,0.f,0.f,0.f,0.f};
      d = wmma_rank1(lo16 ? lx[mt*16+lane15] : 0.f,
                     lo16 ? lx[128+n]        : 0.f, d);
      store_tile<192>(ob, 10864, mt, n, lo16, d);
    } else if (t < 176){                           // p3: x0 (x) x2flat, N=160
      int u=t-96; int mt=u/10, nt=u-mt*10;
      int n = nt*16+lane15;
      v8f d = {0.f,0.f,0.f,0.f,0.f,0.f,0.f,0.f};
      d = wmma_rank1(lo16 ? lx[mt*16+lane15] : 0.f,
                     lo16 ? lx[320+n]        : 0.f, d);
      store_tile<160>(ob, 49120, mt, n, lo16, d);
    } else if (t < 200){                           // p9:  64x96
      int u=t-176; gemm_tile< 96,  0,  35440>(lx, lT, ob, u/6,  u-(u/6)*6,  lane15, lo16);
    } else if (t < 240){                           // p10: 64x160
      int u=t-200; gemm_tile<160, 288, 82640>(lx, lT, ob, u/10, u-(u/10)*10, lane15, lo16);
    } else {                                       // p11: 64x224
      int u=t-240; gemm_tile<224, 768, 92880>(lx, lT, ob, u/14, u-(u/14)*14, lane15, lo16);
    }
  }
}

// =========================== VALU kernel ====================================
// Remaining 15 path regions (pair u<v and diagonal uuu), 39664 elems per row.
__constant__ int   PB[15]  ={0,8128,8256,10272,10336,10832,10864,16912,18400,28480,28800,31280,31440,34912,39376};
__constant__ int   PPAIR[15]={1,0,1,0,1,0,1,1,1,0,1,0,1,1,0};
__constant__ int   PDK[15] ={1,1,1,1,1,1,3,3,5,5,5,5,7,9,9};
__constant__ int   PD1[15] ={1,1,3,3,5,5,3,5,3,3,5,5,5,5,5};
__constant__ int   PMUL[15]={128,128,64,64,32,32,64,32,64,64,32,32,32,32,32};
__constant__ int   PXO[15] ={0,0,128,128,320,320,128,320,128,128,320,320,320,320,320};
__constant__ int   PCG[15] ={0,0,35,35,341,341,44,366,71,71,441,441,566,741,741};
__constant__ float PSC[15] ={1.f,0.5773502692f,1.f,0.4472135955f,1.f,0.3779644730f,1.f,1.f,
                             1.f,0.7071067812f,1.f,0.7071067812f,1.f,1.f,0.7071067812f};
__constant__ int   POUT[15]={0,8128,8256,10272,10336,10832,41584,47632,69600,79680,80000,82480,107216,110688,115152};

__global__ __launch_bounds__(256) void tp_valu_kernel(const float* __restrict__ x,
    const float* __restrict__ cgtab, float* __restrict__ out){
  __shared__ float lx[IN_DIM];
  __shared__ float lcg[CG_FLOATS];
  const int b = blockIdx.y;
  for (int i=threadIdx.x;i<IN_DIM;i+=256)    lx[i]=x[(size_t)b*IN_DIM+i];
  for (int i=threadIdx.x;i<CG_FLOATS;i+=256) lcg[i]=cgtab[i];
  __syncthreads();
  float* ob = out + (size_t)b*OUT_DIM;
  const int e0 = blockIdx.x*1024 + threadIdx.x;
  #pragma unroll
  for (int rep=0; rep<4; ++rep){
    int e = e0 + rep*256;
    if (e >= NVALU) continue;
    int p=0;
    #pragma unroll
    for (int q=1;q<15;++q) if (e >= PB[q]) p=q;
    const int t  = e - PB[p];
    const int dk = PDK[p], d1 = PD1[p], xo = PXO[p];
    const float* cg = &lcg[PCG[p]];
    const int w = t/dk;
    const int k = t - w*dk;
    int u, v;
    if (PPAIR[p]){                    // invert linear pair index -> (u<v)
      const int M = PMUL[p];
      float fm = (float)(2*M-1);
      int uu = (int)(0.5f*(fm - sqrtf(fm*fm - 8.0f*(float)w)));
      if (uu<0) uu=0; if (uu>M-2) uu=M-2;
      while (uu+1<=M-2 && ((uu+1)*(2*M-uu-2)/2 <= w)) ++uu;
      while (uu>0 && (uu*(2*M-uu-1)/2 > w)) --uu;
      u = uu;
      v = u + 1 + (w - u*(2*M-u-1)/2);
    } else { u = w; v = w; }
    float acc=0.f;
    for (int i=0;i<d1;++i){
      float xi = lx[xo+u*d1+i];
      for (int j=0;j<d1;++j)
        acc += xi * lx[xo+v*d1+j] * cg[(i*d1+j)*dk+k];
    }
    __builtin_nontemporal_store(acc*PSC[p], &ob[POUT[p]+t]);
  }
}

// ============================ launcher ======================================
extern "C" void kernel_launch(void* const* d_in, const int* in_sizes, int n_in,
                              void* d_out, int out_size, void* d_ws, size_t ws_size,
                              hipStream_t stream){
  (void)n_in; (void)out_size; (void)ws_size;
  const float* x   = (const float*)d_in[0];
  float*       out = (float*)d_out;
  float*       cg  = (float*)d_ws;                  // 966 floats of CG tables
  const int batch  = in_sizes[0] / IN_DIM;
  hipLaunchKernelGGL(init_cg_kernel, dim3(1), dim3(32), 0, stream, cg);
  hipLaunchKernelGGL(tp_wmma_kernel, dim3(batch), dim3(256), 0, stream, x, cg, out);
  hipLaunchKernelGGL(tp_valu_kernel, dim3((NVALU+1023)/1024, batch), dim3(256), 0, stream,
                     x, cg, out);
}